// Ewald_910533067034
// MI455X (gfx1250) — compile-verified
//
#include <hip/hip_runtime.h>
#include <hip/hip_bf16.h>
#include <math.h>

typedef float v2f __attribute__((ext_vector_type(2)));
typedef float v8f __attribute__((ext_vector_type(8)));

#define NPART 2048
#define NTILE 128                          // NPART / 16
#define NTRI  (NTILE * (NTILE + 1) / 2)    // 8256 lower-triangle tile pairs
#define NBLOCKS 256
#define TPB 256
#define WPB (TPB / 32)                     // 8 waves / block (wave32)
#define TOTAL_WAVES (NBLOCKS * WPB)        // 2048

__global__ __launch_bounds__(TPB) void ewald_pairs_kernel(
    const float* __restrict__ q,
    const float* __restrict__ r,
    const float* __restrict__ u,
    double* __restrict__ partials)
{
    // 64 KB LDS: s_r[i] = (rx,ry,rz,q_i) ; s_u[i] = (ux,uy,uz, r_i·u_i)
    __shared__ __align__(16) unsigned char smem[NPART * 32];
    float4* s_r = (float4*)smem;
    float4* s_u = (float4*)(smem + NPART * 16);

    const int tid = threadIdx.x;

    for (int i = tid; i < NPART; i += TPB) {
        float rx = r[3*i], ry = r[3*i+1], rz = r[3*i+2];
        float ux = u[3*i], uy = u[3*i+1], uz = u[3*i+2];
        s_r[i] = make_float4(rx, ry, rz, q[i]);
        s_u[i] = make_float4(ux, uy, uz, rx*ux + ry*uy + rz*uz);
    }
    __syncthreads();

    const int lane = tid & 31;
    const int half = lane >> 4;    // 0: K=0,1 / rows M,N ; 1: K=2,3 / rows M+8
    const int idx  = lane & 15;
    const int wid  = tid >> 5;
    const int gwave = blockIdx.x * WPB + wid;

    const float A1 = 0.70710678118654752f;   // a = 1/(sigma*sqrt(2))
    const float C1 = 0.79788456080286536f;   // c = 2a/sqrt(pi)
    // Abramowitz-Stegun 7.1.26 erf coefficients (|err| <= 1.5e-7, x >= 0)
    const float PE  = 0.3275911f;
    const float E1  = 0.254829592f;
    const float E2  = -0.284496736f;
    const float E3  = 1.421413741f;
    const float E4  = -1.453152027f;
    const float E5  = 1.061405429f;
    const float NL2E_HALF = -0.72134752044448170368f;  // -0.5 * log2(e)

    double acc = 0.0;

    for (int t = gwave; t < NTRI; t += TOTAL_WAVES) {
        // invert triangular index: ti >= tj
        int ti = (int)((sqrtf(8.0f * (float)t + 1.0f) - 1.0f) * 0.5f);
        while ((ti + 1) * (ti + 2) / 2 <= t) ++ti;
        while (ti * (ti + 1) / 2 > t) --ti;
        const int tj = t - ti * (ti + 1) / 2;

        const int iBase = ti * 16;
        const int jBase = tj * 16;

        const float4 ri4 = s_r[iBase + idx];
        const float4 ui4 = s_u[iBase + idx];
        const float4 rj4 = s_r[jBase + idx];
        const float4 uj4 = s_u[jBase + idx];

        // A (16x4 f32): lanes 0-15 -> {K0,K1}, lanes 16-31 -> {K2,K3}; K3 pad = 0
        // B (4x16 f32) built from j-particle rows with the same per-lane pattern
        v2f aR = { half ? ri4.z : ri4.x, half ? 0.0f : ri4.y };
        v2f aU = { half ? ui4.z : ui4.x, half ? 0.0f : ui4.y };
        v2f bR = { half ? rj4.z : rj4.x, half ? 0.0f : rj4.y };
        v2f bU = { half ? uj4.z : uj4.x, half ? 0.0f : uj4.y };

        v8f cRU = {};  // r_i · u_j
        v8f cUR = {};  // u_i · r_j
        v8f cUU = {};  // u_i · u_j
        cRU = __builtin_amdgcn_wmma_f32_16x16x4_f32(false, aR, false, bU,
                                                    (short)0, cRU, false, false);
        cUR = __builtin_amdgcn_wmma_f32_16x16x4_f32(false, aU, false, bR,
                                                    (short)0, cUR, false, false);
        cUU = __builtin_amdgcn_wmma_f32_16x16x4_f32(false, aU, false, bU,
                                                    (short)0, cUU, false, false);

        const float qj   = rj4.w;   // q_j
        const float rjuj = uj4.w;   // u_j · r_j
        const bool  diag = (ti == tj);

        #pragma unroll
        for (int v = 0; v < 8; ++v) {
            const int M = v + 8 * half;            // C layout: lane n=idx, VGPR v -> M
            const bool valid = (!diag) | (M > idx);
            const int i = iBase + M;

            const float4 rie  = s_r[i];
            const float  riui = ((const float*)&s_u[i])[3];   // u_i · r_i

            // d = r_j - r_i formed directly (no cancellation through Gram matrix)
            const float dx = rj4.x - rie.x;
            const float dy = rj4.y - rie.y;
            const float dz = rj4.z - rie.z;
            float rsq = dx*dx + dy*dy + dz*dz;
            rsq = valid ? rsq : 1.0f;              // keep math finite when masked
            const float qi = rie.w;

            const float riuj = cRU[v];
            const float uirj = cUR[v];
            const float uu   = cUU[v];
            const float dui  = uirj - riui;   // u_i · d
            const float duj  = rjuj - riuj;   // u_j · d

            const float rinv  = __builtin_amdgcn_rsqf(rsq);   // v_rsq_f32
            const float rn    = rsq * rinv;
            const float rinv2 = rinv * rinv;
            const float rinv3 = rinv2 * rinv;

            // gauss = exp(-(a*r)^2) = exp(-rsq/2) via single v_exp_f32
            const float gauss = __builtin_amdgcn_exp2f(NL2E_HALF * rsq);

            // erf(a*rn), x >= 0: 1 - poly(t)*exp(-x^2), t = 1/(1+p*x)
            const float x  = A1 * rn;
            const float tt = __builtin_amdgcn_rcpf(fmaf(PE, x, 1.0f));
            float poly = fmaf(E5, tt, E4);
            poly = fmaf(poly, tt, E3);
            poly = fmaf(poly, tt, E2);
            poly = fmaf(poly, tt, E1);
            poly = poly * tt;
            const float erfv = fmaf(-poly, gauss, 1.0f);

            const float s1 = erfv * rinv3 - C1 * gauss * rinv2;
            const float s2 = 3.0f * erfv * rinv3 - 2.0f * C1 * gauss * rinv2
                           - C1 * gauss;                 // 2*a*a*c == c

            float contrib = qi * qj * erfv * rinv
                          + s1 * (qi * duj - qj * dui)
                          - s2 * rinv2 * dui * duj
                          + s1 * uu;
            contrib = valid ? contrib : 0.0f;      // predicated, no EXEC branch
            acc += (double)contrib;
        }
    }

    // deterministic reduction: wave32 shuffle -> per-block fixed-order sum
    for (int off = 16; off > 0; off >>= 1)
        acc += __shfl_down(acc, off, 32);

    __syncthreads();                       // LDS staging no longer needed
    double* s_red = (double*)smem;
    if (lane == 0) s_red[wid] = acc;
    __syncthreads();
    if (tid == 0) {
        double s = 0.0;
        for (int w = 0; w < WPB; ++w) s += s_red[w];
        partials[blockIdx.x] = s;
    }
}

__global__ void ewald_finalize_kernel(const double* __restrict__ partials,
                                      float* __restrict__ out)
{
    if (threadIdx.x == 0) {
        double s = 0.0;
        for (int b = 0; b < NBLOCKS; ++b) s += partials[b];
        const double SCALE = 90.0474 / 6.283185307179586476925287;  // NORM/(2*pi)
        out[0] = (float)(s * SCALE);
    }
}

extern "C" void kernel_launch(void* const* d_in, const int* in_sizes, int n_in,
                              void* d_out, int out_size, void* d_ws, size_t ws_size,
                              hipStream_t stream)
{
    const float* q = (const float*)d_in[0];   // (2048,)
    const float* r = (const float*)d_in[1];   // (2048,3)
    const float* u = (const float*)d_in[2];   // (2048,3)
    float* out = (float*)d_out;
    double* partials = (double*)d_ws;         // 256 doubles, fully rewritten each call

    ewald_pairs_kernel<<<NBLOCKS, TPB, 0, stream>>>(q, r, u, partials);
    ewald_finalize_kernel<<<1, 32, 0, stream>>>(partials, out);
}